// Attention_5025111736512
// MI455X (gfx1250) — compile-verified
//
#include <hip/hip_runtime.h>
#include <hip/hip_bf16.h>
#include <math.h>

typedef __attribute__((ext_vector_type(2))) float v2f;
typedef __attribute__((ext_vector_type(8))) float v8f;

#define B_  15
#define T_  10
#define H_  4
#define D_  512
#define M_  (B_ * T_)   // 150 rows of the gate GEMM
#define G_  16          // 4*H gates

// ---------------------------------------------------------------------------
// Kernel 1: xg[r, g] = sum_d x[r, d] * W_ih[g, d] + b_ih[g] + b_hh[g]
// M=150 (padded to 160), N=16, K=512 via V_WMMA_F32_16X16X4_F32 (fp32 WMMA).
// One wave (32 lanes) per 16-row tile; 10 blocks total.
//
// Padded rows (150..159) read a clamped (valid) address and produce garbage
// accumulator rows that are simply never stored — no per-element masking
// needed, so loads feed WMMA directly.
// ---------------------------------------------------------------------------
__global__ __launch_bounds__(32)
void lstm_gates_wmma(const float* __restrict__ x,     // [150, 512]
                     const float* __restrict__ Wih,   // [16, 512]
                     const float* __restrict__ bih,   // [16]
                     const float* __restrict__ bhh,   // [16]
                     float* __restrict__ xg)          // [150, 16]
{
  const int lane = threadIdx.x & 31;
  const int mn   = lane & 15;    // M index for A, N index for B/C/D
  const int kv   = lane >> 4;    // K sub-group (0 or 1) within each K=4 step
  const int m0   = blockIdx.x * 16;

  const int arow = m0 + mn;
  // Clamp so OOB lanes load a valid row; their results are never stored.
  const float* __restrict__ arow_p = x + (size_t)(arow < M_ ? arow : 0) * D_;
  const float* __restrict__ brow_p = Wih + (size_t)mn * D_;   // W_ih[g=n][d=k]

  v8f acc0 = {0.f, 0.f, 0.f, 0.f, 0.f, 0.f, 0.f, 0.f};
  v8f acc1 = acc0, acc2 = acc0, acc3 = acc0;

  // k = kb + 2*kv is always even -> 8-byte aligned -> global_load_b64.
#define WSTEP(ACC, KB)                                                        \
  do {                                                                        \
    const int k_ = (KB) + 2 * kv;                                             \
    v2f a_ = *(const v2f*)(arow_p + k_);                                      \
    v2f b_ = *(const v2f*)(brow_p + k_);                                      \
    ACC = __builtin_amdgcn_wmma_f32_16x16x4_f32(false, a_, false, b_,         \
                                                (short)0, ACC, false, false); \
  } while (0)

  // K = 512 -> 128 WMMA steps, four independent accumulator chains for ILP.
#pragma unroll 2
  for (int kb = 0; kb < D_; kb += 16) {
    WSTEP(acc0, kb);
    WSTEP(acc1, kb + 4);
    WSTEP(acc2, kb + 8);
    WSTEP(acc3, kb + 12);
  }
#undef WSTEP

  const float bias = bih[mn] + bhh[mn];
#pragma unroll
  for (int v = 0; v < 8; ++v) {
    const int m = v + 8 * kv;          // C/D layout: lane group selects M half
    const int r = m0 + m;
    if (r < M_) xg[r * G_ + mn] = ((acc0[v] + acc1[v]) + (acc2[v] + acc3[v])) + bias;
  }
}

// ---------------------------------------------------------------------------
// Kernel 2: sequential LSTM scan + attention (faithful to reference bugs).
// One wave; lane b < 15 owns batch b. All state is tiny (H=4, T=10).
// ---------------------------------------------------------------------------
__device__ __forceinline__ float sigmoidf_(float v) {
  return 1.0f / (1.0f + expf(-v));
}

__global__ __launch_bounds__(32)
void lstm_scan_attn(const float* __restrict__ xg,    // [150, 16]
                    const float* __restrict__ h0,    // [1, 15, 4]
                    const float* __restrict__ c0,    // [1, 15, 4]
                    const float* __restrict__ enc,   // [15, 10, 4]
                    const float* __restrict__ Whh,   // [16, 4]
                    float* __restrict__ out)         // [15, 10, 8]
{
  const int b = threadIdx.x;
  if (b >= B_) return;

  float h[H_], c[H_];
#pragma unroll
  for (int j = 0; j < H_; ++j) {
    h[j] = h0[b * H_ + j];
    c[j] = c0[b * H_ + j];
  }

  float att[T_][H_];

  // LSTM scan, torch gate order (i, f, g, o)
  for (int t = 0; t < T_; ++t) {
    const float* xgr = xg + (size_t)(b * T_ + t) * G_;
    float gt[G_];
#pragma unroll
    for (int g = 0; g < G_; ++g) {
      float v = xgr[g];
#pragma unroll
      for (int j = 0; j < H_; ++j) v += h[j] * Whh[g * H_ + j];
      gt[g] = v;
    }
#pragma unroll
    for (int j = 0; j < H_; ++j) {
      const float ig = sigmoidf_(gt[j]);
      const float fg = sigmoidf_(gt[4 + j]);
      const float gg = tanhf(gt[8 + j]);
      const float og = sigmoidf_(gt[12 + j]);
      c[j] = fg * c[j] + ig * gg;
      h[j] = og * tanhf(c[j]);
      att[t][j] = h[j];
    }
  }

  // Attention with faithful reference bugs:
  //  - only the LAST key weight is normalized, and by s**T_IN
  //  - encoder is scaled cumulatively across l (cumprod of w over l)
  const float* encb = enc + (size_t)b * T_ * H_;
  float cumk[T_];
#pragma unroll
  for (int k = 0; k < T_; ++k) cumk[k] = 1.0f;

  for (int l = 0; l < T_; ++l) {
    float ex[T_];
    float s = 0.0f;
    for (int k = 0; k < T_; ++k) {
      float d = 0.0f;
#pragma unroll
      for (int j = 0; j < H_; ++j) d += att[l][j] * encb[k * H_ + j];
      ex[k] = expf(d);
      s += ex[k];
    }
    float sp = 1.0f;
    for (int p = 0; p < T_; ++p) sp *= s;          // s ** T_IN
    ex[T_ - 1] = ex[T_ - 1] / sp;                  // only last key normalized

    for (int k = 0; k < T_; ++k) cumk[k] *= ex[k]; // cumprod over l

    float* o = out + (size_t)(b * T_ + l) * (2 * H_);
#pragma unroll
    for (int j = 0; j < H_; ++j) {
      float cn = 0.0f;
      for (int k = 0; k < T_; ++k) cn += cumk[k] * encb[k * H_ + j];
      o[j]      = cn;        // context
      o[H_ + j] = att[l][j]; // attended hidden state
    }
  }
}

// ---------------------------------------------------------------------------
extern "C" void kernel_launch(void* const* d_in, const int* in_sizes, int n_in,
                              void* d_out, int out_size, void* d_ws, size_t ws_size,
                              hipStream_t stream) {
  const float* x    = (const float*)d_in[0];  // [15,10,512]
  const float* h0   = (const float*)d_in[1];  // [1,15,4]
  const float* c0   = (const float*)d_in[2];  // [1,15,4]
  const float* enc  = (const float*)d_in[3];  // [15,10,4]
  const float* Wih  = (const float*)d_in[4];  // [16,512]
  const float* Whh  = (const float*)d_in[5];  // [16,4]
  const float* bih  = (const float*)d_in[6];  // [16]
  const float* bhh  = (const float*)d_in[7];  // [16]
  float* out = (float*)d_out;                 // [15,10,8]

  float* xg = (float*)d_ws;                   // [150,16] = 9600 B scratch

  lstm_gates_wmma<<<dim3((M_ + 15) / 16), dim3(32), 0, stream>>>(x, Wih, bih, bhh, xg);
  lstm_scan_attn<<<dim3(1), dim3(32), 0, stream>>>(xg, h0, c0, enc, Whh, out);
}